// CosmicNetGNN_4123168604820
// MI455X (gfx1250) — compile-verified
//
#include <hip/hip_runtime.h>

// ---------------------------------------------------------------------------
// Problem constants (from reference)
// ---------------------------------------------------------------------------
#define N_NODES 16384
#define N_EDGES 65536
#define NGRAPH  32
#define H       64
#define LAYERS  3
#define NEGS    0.1f
#define LN_EPS  1e-5f

// Extended K for the fused big GEMM: 64*64 outer-product columns + 64 bias cols
#define KEXT    4160              // 4096 + 64
#define NCHUNK  (KEXT / 4)        // 1040 K-chunks of 4 (f32 WMMA K=4)
// Packed B ("Bm2") layout per layer: [oblk(4)][ichunk(65)][kq(16)][lane(32)][2]
#define BM2_PER_OBLK  (65 * 16 * 32 * 2)   // 66560 floats
#define BM2_PER_LAYER (4 * BM2_PER_OBLK)   // 266240 floats

typedef __attribute__((ext_vector_type(2))) float vf2;
typedef __attribute__((ext_vector_type(8))) float vf8;

__device__ __forceinline__ float lrelu(float v) { return v > 0.f ? v : NEGS * v; }

// ---------------------------------------------------------------------------
// zero fill
// ---------------------------------------------------------------------------
__global__ void k_zero(float* __restrict__ p, int n) {
    int g = blockIdx.x * blockDim.x + threadIdx.x;
    if (g < n) p[g] = 0.f;
}

// ---------------------------------------------------------------------------
// in-degree (by dst) and per-graph node counts
// ---------------------------------------------------------------------------
__global__ void k_degcnt(const int* __restrict__ dst, const int* __restrict__ batch,
                         float* __restrict__ deg, float* __restrict__ cnt) {
    int g = blockIdx.x * blockDim.x + threadIdx.x;
    if (g < N_EDGES) atomicAdd(&deg[dst[g]], 1.f);
    if (g < N_NODES) atomicAdd(&cnt[batch[g]], 1.f);
}

// ---------------------------------------------------------------------------
// input projection: h = leaky(x @ W_in.T + b_in)   [N,4] -> [N,64]
// ---------------------------------------------------------------------------
__global__ void k_inproj(const float* __restrict__ x, const float* __restrict__ W,
                         const float* __restrict__ b, float* __restrict__ h) {
    int g = blockIdx.x * blockDim.x + threadIdx.x;   // N*H threads
    int n = g >> 6, o = g & 63;
    float acc = b[o];
#pragma unroll
    for (int k = 0; k < 4; ++k) acc += x[n * 4 + k] * W[o * 4 + k];
    h[g] = lrelu(acc);
}

// ---------------------------------------------------------------------------
// Repack ew3/eb3 (all layers) into WMMA-fragment-ordered B matrix.
//   BigB[kk][o] = (kk<4096) ? ew3[(i*64+o)*64 + k]  (i=kk>>6, k=kk&63)
//               :            eb3[(kk-4096)*64 + o]
//   Bm2 flat idx = ((((l*4+oblk)*65 + ic)*16 + kq)*32 + lane)*2 + t
//   with kk = ic*64 + kq*4 + 2*(lane>>4) + t,  o = oblk*16 + (lane&15)
// ---------------------------------------------------------------------------
__global__ void k_prepB(const float* __restrict__ ew3, const float* __restrict__ eb3,
                        float* __restrict__ Bm2) {
    int idx = blockIdx.x * blockDim.x + threadIdx.x;
    if (idx >= LAYERS * BM2_PER_LAYER) return;
    int t    = idx & 1;
    int lane = (idx >> 1) & 31;
    int kq   = (idx >> 6) & 15;
    int q    = idx >> 10;            // (l*4 + oblk)*65 + ic
    int ic   = q % 65;
    int p    = q / 65;
    int oblk = p & 3;
    int l    = p >> 2;
    int kk = ic * 64 + kq * 4 + ((lane >> 4) << 1) + t;
    int o  = oblk * 16 + (lane & 15);
    float v;
    if (kk < 4096) {
        int i = kk >> 6, k = kk & 63;
        v = ew3[(size_t)l * 4096 * 64 + (size_t)(i * 64 + o) * 64 + k];
    } else {
        v = eb3[l * 4096 + (kk - 4096) * 64 + o];
    }
    Bm2[idx] = v;
}

// ---------------------------------------------------------------------------
// edge MLP stages 1+2: e2 = leaky(leaky(ea@w1.T+b1)@w2.T+b2)   [E,5] -> [E,64]
// block = 256 threads = 4 edges x 64 outs
// ---------------------------------------------------------------------------
__global__ void k_edgemlp(const float* __restrict__ eattr,
                          const float* __restrict__ w1, const float* __restrict__ b1,
                          const float* __restrict__ w2, const float* __restrict__ b2,
                          float* __restrict__ e2out) {
    __shared__ float e1s[4][64];
    int tid = threadIdx.x;
    int el = tid >> 6, o = tid & 63;
    int e = blockIdx.x * 4 + el;
    float acc = b1[o];
#pragma unroll
    for (int k = 0; k < 5; ++k) acc += eattr[e * 5 + k] * w1[o * 5 + k];
    e1s[el][o] = lrelu(acc);
    __syncthreads();
    acc = b2[o];
#pragma unroll 8
    for (int i = 0; i < 64; ++i) acc += e1s[el][i] * w2[o * 64 + i];
    e2out[(size_t)e * 64 + o] = lrelu(acc);
}

// ---------------------------------------------------------------------------
// MAIN fused NNConv message kernel (WMMA f32 16x16x4).
// Grid: E/64 blocks, 256 threads (8 waves). Block processes 64 edges.
// Wave w: out-block = w&3 (16 outs), edge subtiles {2*(w>>2), 2*(w>>2)+1}.
// A[e, i*64+k] = h_src[e][i]*e2[e][k];  A[e, 4096+j] = h_src[e][j].
// Accumulate msg, scatter-add into agg[dst] (mean divide happens later).
// ---------------------------------------------------------------------------
#define LSTR 66   // padded LDS row stride (8B aligned, conflict-light)

__global__ void __launch_bounds__(256) k_gnn_msg(
        const float* __restrict__ h, const float* __restrict__ e2,
        const float* __restrict__ Bm2l, const int* __restrict__ src,
        const int* __restrict__ dst, float* __restrict__ agg) {
    __shared__ float hT[64 * LSTR];
    __shared__ float e2T[64 * LSTR];
    const int tid = threadIdx.x;
    const int etb = blockIdx.x * 64;

    // cooperative gather of h[src] tile and e2 tile (64 edges x 64 feats)
    for (int t = tid; t < 64 * 64; t += 256) {
        int er = t >> 6, c = t & 63;
        int e = etb + er;
        hT[er * LSTR + c]  = h[(size_t)src[e] * 64 + c];
        e2T[er * LSTR + c] = e2[(size_t)e * 64 + c];
    }
    __syncthreads();

    const int w = tid >> 5, lane = tid & 31;
    const int oblk = w & 3;
    const int s0 = (w >> 2) * 2;
    const int m = lane & 15;
    const int kh2 = (lane >> 4) << 1;          // 0 or 2: K-pair offset within chunk
    const int r0 = s0 * 16 + m, r1 = r0 + 16;  // A rows for the two subtiles
    const float2* __restrict__ bp = (const float2*)(Bm2l + oblk * BM2_PER_OBLK);

    vf8 c0 = {0.f, 0.f, 0.f, 0.f, 0.f, 0.f, 0.f, 0.f};
    vf8 c1 = {0.f, 0.f, 0.f, 0.f, 0.f, 0.f, 0.f, 0.f};

    // outer-product K blocks: i = h-feature index, inner k = e2-feature index
    for (int i = 0; i < 64; ++i) {
        const float hv0 = hT[r0 * LSTR + i];
        const float hv1 = hT[r1 * LSTR + i];
        const float2* __restrict__ bpi = bp + (size_t)i * 16 * 32 + lane;
#pragma unroll
        for (int kq = 0; kq < 16; ++kq) {
            const float2 bv = bpi[kq * 32];
            const int kb = kq * 4 + kh2;
            vf2 a0, a1, bb;
            a0.x = hv0 * e2T[r0 * LSTR + kb];
            a0.y = hv0 * e2T[r0 * LSTR + kb + 1];
            a1.x = hv1 * e2T[r1 * LSTR + kb];
            a1.y = hv1 * e2T[r1 * LSTR + kb + 1];
            bb.x = bv.x; bb.y = bv.y;
            c0 = __builtin_amdgcn_wmma_f32_16x16x4_f32(false, a0, false, bb,
                                                       (short)0, c0, false, false);
            c1 = __builtin_amdgcn_wmma_f32_16x16x4_f32(false, a1, false, bb,
                                                       (short)0, c1, false, false);
        }
    }
    // bias K block: A[m, 4096+j] = h[m][j]
    {
        const float2* __restrict__ bpi = bp + (size_t)64 * 16 * 32 + lane;
#pragma unroll
        for (int kq = 0; kq < 16; ++kq) {
            const float2 bv = bpi[kq * 32];
            const int kb = kq * 4 + kh2;
            vf2 a0, a1, bb;
            a0.x = hT[r0 * LSTR + kb];
            a0.y = hT[r0 * LSTR + kb + 1];
            a1.x = hT[r1 * LSTR + kb];
            a1.y = hT[r1 * LSTR + kb + 1];
            bb.x = bv.x; bb.y = bv.y;
            c0 = __builtin_amdgcn_wmma_f32_16x16x4_f32(false, a0, false, bb,
                                                       (short)0, c0, false, false);
            c1 = __builtin_amdgcn_wmma_f32_16x16x4_f32(false, a1, false, bb,
                                                       (short)0, c1, false, false);
        }
    }

    // scatter-add: D layout — VGPR r, lanes 0-15: M=r, N=lane; lanes 16-31: M=8+r
    const int n = lane & 15;
    const int mh = lane >> 4;
#pragma unroll
    for (int r = 0; r < 8; ++r) {
        const int e0 = etb + s0 * 16 + mh * 8 + r;
        const int e1 = e0 + 16;
        atomicAdd(&agg[(size_t)dst[e0] * 64 + oblk * 16 + n], c0[r]);
        atomicAdd(&agg[(size_t)dst[e1] * 64 + oblk * 16 + n], c1[r]);
    }
}

// ---------------------------------------------------------------------------
// node update: out = agg/deg + h@root_w.T + root_b; LayerNorm; h' = leaky+res
// block = 256 = 8 waves, one node per wave, each lane owns outs {lane, lane+32}
// ---------------------------------------------------------------------------
__global__ void k_node(const float* __restrict__ hc, const float* __restrict__ agg,
                       const float* __restrict__ deg, const float* __restrict__ rw,
                       const float* __restrict__ rb, const float* __restrict__ lng,
                       const float* __restrict__ lnb, float* __restrict__ hn) {
    const int wv = threadIdx.x >> 5, lane = threadIdx.x & 31;
    const int node = blockIdx.x * 8 + wv;
    const int o0 = lane, o1 = lane + 32;
    float acc0 = rb[o0], acc1 = rb[o1];
#pragma unroll 8
    for (int i = 0; i < 64; ++i) {
        const float hv = hc[(size_t)node * 64 + i];
        acc0 += hv * rw[o0 * 64 + i];
        acc1 += hv * rw[o1 * 64 + i];
    }
    const float dv = fmaxf(deg[node], 1.f);
    float a0 = agg[(size_t)node * 64 + o0] / dv + acc0;
    float a1 = agg[(size_t)node * 64 + o1] / dv + acc1;
    float s = a0 + a1;
    for (int off = 16; off > 0; off >>= 1) s += __shfl_xor(s, off, 32);
    const float mu = s * (1.f / 64.f);
    const float d0 = a0 - mu, d1 = a1 - mu;
    float v = d0 * d0 + d1 * d1;
    for (int off = 16; off > 0; off >>= 1) v += __shfl_xor(v, off, 32);
    const float rs = rsqrtf(v * (1.f / 64.f) + LN_EPS);
    const float y0 = d0 * rs * lng[o0] + lnb[o0];
    const float y1 = d1 * rs * lng[o1] + lnb[o1];
    hn[(size_t)node * 64 + o0] = lrelu(y0) + hc[(size_t)node * 64 + o0];
    hn[(size_t)node * 64 + o1] = lrelu(y1) + hc[(size_t)node * 64 + o1];
}

// ---------------------------------------------------------------------------
// global mean pool (sum via atomics; divide in head)
// ---------------------------------------------------------------------------
__global__ void k_pool(const float* __restrict__ h, const int* __restrict__ batch,
                       float* __restrict__ pooled) {
    int g = blockIdx.x * blockDim.x + threadIdx.x;   // N*H
    int nd = g >> 6, o = g & 63;
    atomicAdd(&pooled[batch[nd] * 64 + o], h[g]);
}

// ---------------------------------------------------------------------------
// prediction head (single block, tiny)
// ---------------------------------------------------------------------------
__global__ void k_head(const float* __restrict__ pooled, const float* __restrict__ cnt,
                       const float* __restrict__ pw1, const float* __restrict__ pb1,
                       const float* __restrict__ pw2, const float* __restrict__ pb2,
                       const float* __restrict__ pw3, const float* __restrict__ pb3,
                       float* __restrict__ out) {
    __shared__ float pm[NGRAPH * 64];
    __shared__ float p1[NGRAPH * 64];
    __shared__ float p2[NGRAPH * 32];
    const int tid = threadIdx.x;
    for (int idx = tid; idx < NGRAPH * 64; idx += 256) {
        int g = idx >> 6;
        pm[idx] = pooled[idx] / fmaxf(cnt[g], 1.f);
    }
    __syncthreads();
    for (int idx = tid; idx < NGRAPH * 64; idx += 256) {
        int g = idx >> 6, o = idx & 63;
        float acc = pb1[o];
        for (int i = 0; i < 64; ++i) acc += pm[g * 64 + i] * pw1[o * 64 + i];
        p1[idx] = lrelu(acc);
    }
    __syncthreads();
    for (int idx = tid; idx < NGRAPH * 32; idx += 256) {
        int g = idx >> 5, j = idx & 31;
        float acc = pb2[j];
        for (int i = 0; i < 64; ++i) acc += p1[g * 64 + i] * pw2[j * 64 + i];
        p2[idx] = lrelu(acc);
    }
    __syncthreads();
    if (tid < NGRAPH) {
        float acc = pb3[0];
        for (int j = 0; j < 32; ++j) acc += p2[tid * 32 + j] * pw3[j];
        out[tid] = acc;
    }
}

// ---------------------------------------------------------------------------
// launch
// ---------------------------------------------------------------------------
extern "C" void kernel_launch(void* const* d_in, const int* in_sizes, int n_in,
                              void* d_out, int out_size, void* d_ws, size_t ws_size,
                              hipStream_t stream) {
    (void)in_sizes; (void)n_in; (void)out_size; (void)ws_size;
    const float* x      = (const float*)d_in[0];
    const int*   eidx   = (const int*)  d_in[1];
    const float* eattr  = (const float*)d_in[2];
    const int*   batch  = (const int*)  d_in[3];
    const float* W_in   = (const float*)d_in[4];
    const float* b_in   = (const float*)d_in[5];
    const float* ew1    = (const float*)d_in[6];
    const float* eb1    = (const float*)d_in[7];
    const float* ew2    = (const float*)d_in[8];
    const float* eb2    = (const float*)d_in[9];
    const float* ew3    = (const float*)d_in[10];
    const float* eb3    = (const float*)d_in[11];
    const float* root_w = (const float*)d_in[12];
    const float* root_b = (const float*)d_in[13];
    const float* ln_g   = (const float*)d_in[14];
    const float* ln_b   = (const float*)d_in[15];
    const float* pw1    = (const float*)d_in[16];
    const float* pb1    = (const float*)d_in[17];
    const float* pw2    = (const float*)d_in[18];
    const float* pb2    = (const float*)d_in[19];
    const float* pw3    = (const float*)d_in[20];
    const float* pb3    = (const float*)d_in[21];
    float* out = (float*)d_out;

    const int* src = eidx;
    const int* dst = eidx + N_EDGES;

    // workspace carve-up (floats)
    float* ws = (float*)d_ws;
    size_t off = 0;
    float* hA     = ws + off; off += (size_t)N_NODES * H;
    float* hB     = ws + off; off += (size_t)N_NODES * H;
    float* e2b    = ws + off; off += (size_t)N_EDGES * H;
    float* Bm2    = ws + off; off += (size_t)LAYERS * BM2_PER_LAYER;
    float* agg    = ws + off; off += (size_t)N_NODES * H;
    float* deg    = ws + off; off += N_NODES;
    float* cnt    = ws + off; off += NGRAPH;
    float* pooled = ws + off; off += NGRAPH * H;

    // zero deg/cnt/pooled
    {
        int n = N_NODES + NGRAPH + NGRAPH * H;
        k_zero<<<(n + 255) / 256, 256, 0, stream>>>(deg, n);  // contiguous region
    }
    k_degcnt<<<N_EDGES / 256, 256, 0, stream>>>(dst, batch, deg, cnt);
    k_inproj<<<(N_NODES * H) / 256, 256, 0, stream>>>(x, W_in, b_in, hA);
    k_prepB<<<(LAYERS * BM2_PER_LAYER) / 256, 256, 0, stream>>>(ew3, eb3, Bm2);

    const float* hc = hA;
    float* hn = hB;
    for (int l = 0; l < LAYERS; ++l) {
        k_edgemlp<<<N_EDGES / 4, 256, 0, stream>>>(
            eattr, ew1 + (size_t)l * H * 5, eb1 + l * H,
            ew2 + (size_t)l * H * H, eb2 + l * H, e2b);
        k_zero<<<(N_NODES * H) / 256, 256, 0, stream>>>(agg, N_NODES * H);
        k_gnn_msg<<<N_EDGES / 64, 256, 0, stream>>>(
            hc, e2b, Bm2 + (size_t)l * BM2_PER_LAYER, src, dst, agg);
        k_node<<<N_NODES / 8, 256, 0, stream>>>(
            hc, agg, deg, root_w + (size_t)l * H * H, root_b + l * H,
            ln_g + l * H, ln_b + l * H, hn);
        const float* tmp = hc; hc = hn; hn = (float*)tmp;
    }

    k_pool<<<(N_NODES * H) / 256, 256, 0, stream>>>(hc, batch, pooled);
    k_head<<<1, 256, 0, stream>>>(pooled, cnt, pw1, pb1, pw2, pb2, pw3, pb3, out);
}